// CumulantNN_1005022347603
// MI455X (gfx1250) — compile-verified
//
#include <hip/hip_runtime.h>
#include <math.h>

// Problem constants (match reference)
static constexpr int    NSPIN  = 8192;
static constexpr int    NB_    = 5;
static constexpr float  TWOPI_ = 6.28318530717958647692f;

// K-octet split for L2 residency: ktp < KRES is L2-resident (RT loads),
// ktp >= KRES streams from HBM with non-temporal loads.
// Resident bytes = 512 row-tiles * KRES * 512B = 160 MB  (< 192 MB L2)
static constexpr int    KRES   = 640;         // of 1024 K-octets
static constexpr int    KTPT   = NSPIN / 8;   // 1024

typedef float v2f __attribute__((ext_vector_type(2)));
typedef float v4f __attribute__((ext_vector_type(4)));   // native vector: OK for nontemporal builtins
typedef float v8f __attribute__((ext_vector_type(8)));

__device__ __forceinline__ float sig2m1(float x) {
    // 2*sigmoid(x) - 1, matching the reference formula in fp32
    return 2.0f / (1.0f + expf(-x)) - 1.0f;
}

// -----------------------------------------------------------------------------
// Kernel 1: Jeff = 2*sigmoid(Js)-1, repacked into WMMA A-fragment order.
//
// Fragment layout for V_WMMA_F32_16X16X4_F32 A-matrix (16x4, fp32, ISA 7.12.2):
//   lane = (M % 16) + 16*half,  VGPR v holds K = 2*half + v   (v = 0,1)
// We store, per (row-tile mt, K-octet ktp), one 16B slot per lane:
//   slot dwords = { J[r][k0+2h], J[r][k0+2h+1], J[r][k0+4+2h], J[r][k0+5+2h] }
//   r = mt*16 + lane%16, h = lane/16, k0 = ktp*8
// Slots for consecutive ktp are contiguous -> the GEMV streams coalesced b128s.
// The streaming region (ktp >= KRES) is written non-temporally to avoid
// polluting the L2 slice we want to keep resident.
// -----------------------------------------------------------------------------
__global__ void __launch_bounds__(256) jeff_pack_kernel(
    const float* __restrict__ Js, float* __restrict__ Jf)
{
    __shared__ float tile[16][128];
    const int t  = threadIdx.x;          // 0..255
    const int mt = blockIdx.x;           // 0..511  (16-row tile)
    const int cb = blockIdx.y;           // 0..63   (128-col block = 16 ktp)
    const int c0 = cb * 128;

    // Load 16x128 tile coalesced (8 floats/thread), apply 2*sigmoid-1
    {
        const int r = t >> 5;            // 0..7
        const int c = (t & 31) * 4;      // 0..124
        const v4f a = *(const v4f*)(Js + (size_t)(mt * 16 + r)     * NSPIN + c0 + c);
        const v4f b = *(const v4f*)(Js + (size_t)(mt * 16 + r + 8) * NSPIN + c0 + c);
        tile[r][c + 0] = sig2m1(a.x); tile[r][c + 1] = sig2m1(a.y);
        tile[r][c + 2] = sig2m1(a.z); tile[r][c + 3] = sig2m1(a.w);
        tile[r + 8][c + 0] = sig2m1(b.x); tile[r + 8][c + 1] = sig2m1(b.y);
        tile[r + 8][c + 2] = sig2m1(b.z); tile[r + 8][c + 3] = sig2m1(b.w);
    }
    __syncthreads();

    // Write 512 fragment slots (16 ktp * 32 lanes), fully coalesced 16B stores
    for (int s = t; s < 512; s += 256) {
        const int ktpl = s >> 5;         // 0..15
        const int lane = s & 31;
        const int r    = lane & 15;
        const int h    = lane >> 4;
        const int kl   = ktpl * 8 + 2 * h;
        v4f o;
        o.x = tile[r][kl];     o.y = tile[r][kl + 1];
        o.z = tile[r][kl + 4]; o.w = tile[r][kl + 5];
        const size_t ktp  = (size_t)cb * 16 + ktpl;
        const size_t slot = ((size_t)mt * KTPT + ktp) * 32 + lane;
        v4f* dst = (v4f*)Jf + slot;
        if ((int)ktp >= KRES) {
            __builtin_nontemporal_store(o, dst);   // streaming region: don't cache
        } else {
            *dst = o;                              // resident region: normal RT
        }
    }
}

// -----------------------------------------------------------------------------
// Kernel 2: one Euler step.
//   h = Jeff @ sz          (fp32 WMMA GEMV, sz broadcast across B columns)
//   u = vs @ cos(2*pi*t*nb),  pt = 2*sigmoid(u)-1
//   state update, written to the other buffer.
// Block = 64 threads (2 waves); wave w owns row-tile mt = blockIdx.x*2 + w.
// -----------------------------------------------------------------------------
__device__ __forceinline__ void gemv_mac(const v4f a, const int k0, const int bo,
                                         const float* s_sz, v8f& acc0, v8f& acc1)
{
    v2f a0 = {a.x, a.y};
    v2f a1 = {a.z, a.w};
    // B fragments: sz[k] broadcast across all 16 columns (K = 2*half + v)
    v2f b0 = {s_sz[k0 + bo],     s_sz[k0 + bo + 1]};
    v2f b1 = {s_sz[k0 + 4 + bo], s_sz[k0 + 5 + bo]};
    acc0 = __builtin_amdgcn_wmma_f32_16x16x4_f32(false, a0, false, b0,
                                                 (short)0, acc0, false, false);
    acc1 = __builtin_amdgcn_wmma_f32_16x16x4_f32(false, a1, false, b1,
                                                 (short)0, acc1, false, false);
}

__global__ void __launch_bounds__(64) spin_step_kernel(
    const float* __restrict__ Jf, const float* __restrict__ vs,
    const float* __restrict__ times, const float* __restrict__ xin,
    float* __restrict__ xout, int step)
{
    __shared__ float s_sz[NSPIN];        // 32 KB: full sz vector
    __shared__ float s_h[32];            // h for this block's 32 rows

    const int tid = threadIdx.x;
    for (int i = tid; i < NSPIN; i += 64) s_sz[i] = xin[2 * NSPIN + i];
    __syncthreads();

    const int wave = tid >> 5;
    const int lane = tid & 31;
    const int half = lane >> 4;
    const int bo   = 2 * half;                       // K base offset for this lane-half
    const int mt   = blockIdx.x * 2 + wave;          // 0..511

    v8f acc0 = {0.f, 0.f, 0.f, 0.f, 0.f, 0.f, 0.f, 0.f};
    v8f acc1 = {0.f, 0.f, 0.f, 0.f, 0.f, 0.f, 0.f, 0.f};

    const v4f* Jp = (const v4f*)Jf + (size_t)mt * KTPT * 32 + lane;

    // L2-resident portion: regular (RT) loads -> stays cached across all steps
#pragma unroll 4
    for (int ktp = 0; ktp < KRES; ++ktp) {
        const v4f a = Jp[(size_t)ktp * 32];          // coalesced 512B per wave
        gemv_mac(a, ktp * 8, bo, s_sz, acc0, acc1);
    }
    // Streaming portion: non-temporal loads -> evict-first, preserves resident set
#pragma unroll 4
    for (int ktp = KRES; ktp < KTPT; ++ktp) {
        const v4f a = __builtin_nontemporal_load(Jp + (size_t)ktp * 32);
        gemv_mac(a, ktp * 8, bo, s_sz, acc0, acc1);
    }

    v8f hs = acc0 + acc1;   // D replicated across N; VGPR v holds row v + 8*half

    if ((lane & 15) == 0) {
#pragma unroll
        for (int v = 0; v < 8; ++v)
            s_h[wave * 16 + half * 8 + v] = hs[v];
    }
    __syncthreads();

    // Per-spin update for this block's 32 rows
    if (tid < 32) {
        const int   r  = blockIdx.x * 32 + tid;
        const float t0 = times[step];
        const float dt = times[step + 1] - t0;

        float u = 0.f;
#pragma unroll
        for (int b = 0; b < NB_; ++b)
            u += vs[(size_t)r * NB_ + b] * cosf(TWOPI_ * t0 * (float)b);
        const float pt = 2.0f / (1.0f + expf(-u)) - 1.0f;

        const float h  = s_h[tid];
        const float sx = xin[r];
        const float sy = xin[NSPIN + r];
        const float sz = s_sz[r];

        xout[r]             = sx + dt * (-2.0f * h * sy);
        xout[NSPIN + r]     = sy + dt * ( 2.0f * h * sx - 2.0f * pt * sz);
        xout[2 * NSPIN + r] = sz + dt * ( 2.0f * pt * sy);
    }
}

// -----------------------------------------------------------------------------
// Launcher. ws layout: [ Jf : N*N fp32 (256 MB) | buf0 : 3N | buf1 : 3N ]
// inputs (setup_inputs order): times(65), Js(N*N), vs(N*NB), x0(3N)
// -----------------------------------------------------------------------------
extern "C" void kernel_launch(void* const* d_in, const int* in_sizes, int n_in,
                              void* d_out, int out_size, void* d_ws, size_t ws_size,
                              hipStream_t stream) {
    (void)in_sizes; (void)n_in; (void)out_size; (void)ws_size;
    const float* times = (const float*)d_in[0];
    const float* Js    = (const float*)d_in[1];
    const float* vs    = (const float*)d_in[2];
    const float* x0    = (const float*)d_in[3];

    float* Jf   = (float*)d_ws;
    float* buf0 = (float*)((char*)d_ws + (size_t)NSPIN * NSPIN * sizeof(float));
    float* buf1 = buf0 + 3 * NSPIN;
    float* out  = (float*)d_out;

    // One-time (per launch) Jeff precompute + fragment pack
    dim3 pgrid(NSPIN / 16, NSPIN / 128);   // 512 x 64
    jeff_pack_kernel<<<pgrid, 256, 0, stream>>>(Js, Jf);

    // 64 sequential Euler steps, double-buffered; last step writes d_out
    for (int s = 0; s < 64; ++s) {
        const float* xin  = (s == 0)  ? x0  : ((s & 1) ? buf0 : buf1);
        float*       xout = (s == 63) ? out : ((s & 1) ? buf1 : buf0);
        spin_step_kernel<<<dim3(NSPIN / 32), 64, 0, stream>>>(Jf, vs, times, xin, xout, s);
    }
}